// Net_38998303048474
// MI455X (gfx1250) — compile-verified
//
#include <hip/hip_runtime.h>
#include <hip/hip_bf16.h>
#include <math.h>

// ---------------------------------------------------------------------------
// 3-layer GCN (PyG gcn_norm w/ self-loops), f32 end-to-end.
//   deg[c] = 1 + #edges into c ; dinv = rsqrt(deg)
//   per layer: h = act @ W (WMMA f32 16x16x4)
//              acc[i] = h[i]*dinv[i]^2  (self-loop)
//              acc[col[e]] += h[row[e]] * dinv[row]*dinv[col]  (f32 atomics, L2)
//              act = relu(acc + b)   (final layer: L2 row-normalize)
// All feature matrices (25.6 MB) are L2-resident on MI455X (192 MB L2).
// ---------------------------------------------------------------------------

typedef __attribute__((ext_vector_type(2))) float v2f;
typedef __attribute__((ext_vector_type(8))) float v8f;

// ---- degree / norm kernels -------------------------------------------------
__global__ void k_set1(float* d, int N) {
    int i = blockIdx.x * blockDim.x + threadIdx.x;
    if (i < N) d[i] = 1.0f;  // self-loop contributes 1 to every degree
}

__global__ void k_deg(const int* __restrict__ col, float* deg, int E) {
    int e = blockIdx.x * blockDim.x + threadIdx.x;
    if (e < E) atomicAdd(&deg[col[e]], 1.0f);
}

__global__ void k_dinv(float* d, int N) {
    int i = blockIdx.x * blockDim.x + threadIdx.x;
    if (i < N) d[i] = 1.0f / sqrtf(d[i]);  // deg >= 1 always (self-loops)
}

// ---- GEMM: H[N,M] = X[N,K] @ W[K,M], row-major, via V_WMMA_F32_16X16X4_F32 -
// One wave computes one 16x16 output tile; K-loop in steps of 4.
// A frag (16x4 f32): lanes 0-15 hold K=k0,k0+1 ; lanes 16-31 hold K=k0+2,k0+3.
// B frag (4x16 f32): lane n holds B[k][col0+n]; lanes 16-31 carry K=k0+2,k0+3.
// C/D  (16x16 f32): VGPR r = row r (lanes 0-15) / row 8+r (lanes 16-31).
__global__ void __launch_bounds__(128)
k_gemm_f32_wmma(const float* __restrict__ X, const float* __restrict__ W,
                float* __restrict__ H, int N, int K, int M) {
    const int waveId = threadIdx.x >> 5;
    const int lane   = threadIdx.x & 31;
    const int numColTiles = M >> 4;
    const int numRowTiles = N >> 4;
    const int tile = blockIdx.x * 4 + waveId;          // 4 waves / block
    if (tile >= numRowTiles * numColTiles) return;      // wave-uniform branch
    const int row0 = (tile / numColTiles) << 4;
    const int col0 = (tile % numColTiles) << 4;
    const int half = lane >> 4;        // 0: lanes 0-15, 1: lanes 16-31
    const int l    = lane & 15;

    v8f c = {0.f, 0.f, 0.f, 0.f, 0.f, 0.f, 0.f, 0.f};
    for (int k0 = 0; k0 < K; k0 += 4) {
        const int kk = k0 + (half << 1);
        const float* ap = X + (size_t)(row0 + l) * K + kk;   // contiguous pair
        v2f a; a.x = ap[0]; a.y = ap[1];
        const float* bp = W + (size_t)kk * M + col0 + l;
        v2f b; b.x = bp[0]; b.y = bp[M];                     // rows kk, kk+1
        c = __builtin_amdgcn_wmma_f32_16x16x4_f32(
                /*neg_a=*/false, a, /*neg_b=*/false, b,
                /*c_mod=*/(short)0, c, /*reuse_a=*/false, /*reuse_b=*/false);
    }
    float* out = H + (size_t)(row0 + (half << 3)) * M + col0 + l;
#pragma unroll
    for (int r = 0; r < 8; ++r) out[(size_t)r * M] = c[r];
}

// Scalar fallback for row tail (N % 16 != 0). For N=100000 this never runs.
__global__ void k_gemm_tail(const float* __restrict__ X, const float* __restrict__ W,
                            float* __restrict__ H, int rowStart, int N, int K, int M) {
    int idx = blockIdx.x * blockDim.x + threadIdx.x;
    int nRows = N - rowStart;
    if (idx >= nRows * M) return;
    int r = rowStart + idx / M, m = idx % M;
    float s = 0.f;
    for (int k = 0; k < K; ++k) s = fmaf(X[(size_t)r * K + k], W[(size_t)k * M + m], s);
    H[(size_t)r * M + m] = s;
}

// ---- aggregation -----------------------------------------------------------
__global__ void k_selfloop(const float* __restrict__ h, const float* __restrict__ dinv,
                           float* __restrict__ acc, int N, int d) {
    size_t i = (size_t)blockIdx.x * blockDim.x + threadIdx.x;
    if (i >= (size_t)N * d) return;
    int node = (int)(i / d);
    float w = dinv[node];
    acc[i] = h[i] * w * w;
}

// One thread per (edge, 4-channel group): float4 gather + 4 f32 atomics.
__global__ void k_scatter(const float* __restrict__ h, const int* __restrict__ row,
                          const int* __restrict__ col, const float* __restrict__ dinv,
                          float* acc, int E, int d) {
    const int dg = d >> 2;
    size_t tid = (size_t)blockIdx.x * blockDim.x + threadIdx.x;
    if (tid >= (size_t)E * dg) return;
    int e  = (int)(tid / dg);
    int cg = (int)(tid % dg);
    int r = row[e], c = col[e];
    float w = dinv[r] * dinv[c];
    const float4 hv = *reinterpret_cast<const float4*>(h + (size_t)r * d + cg * 4);
    float* ap = acc + (size_t)c * d + cg * 4;
    atomicAdd(ap + 0, hv.x * w);
    atomicAdd(ap + 1, hv.y * w);
    atomicAdd(ap + 2, hv.z * w);
    atomicAdd(ap + 3, hv.w * w);
}

__global__ void k_bias_relu(const float* __restrict__ acc, const float* __restrict__ b,
                            float* __restrict__ out, int N, int d) {
    size_t i = (size_t)blockIdx.x * blockDim.x + threadIdx.x;
    if (i >= (size_t)N * d) return;
    int c = (int)(i % d);
    out[i] = fmaxf(acc[i] + b[c], 0.0f);
}

// One wave32 per row (d = 64, 2 floats per lane), shuffle reduction.
__global__ void __launch_bounds__(128)
k_l2norm(const float* __restrict__ acc, const float* __restrict__ b,
         float* __restrict__ out, int N) {
    int wave = (int)(((size_t)blockIdx.x * blockDim.x + threadIdx.x) >> 5);
    int lane = threadIdx.x & 31;
    if (wave >= N) return;
    const float* ar = acc + (size_t)wave * 64 + lane * 2;
    float e0 = ar[0] + b[lane * 2];
    float e1 = ar[1] + b[lane * 2 + 1];
    float ss = e0 * e0 + e1 * e1;
#pragma unroll
    for (int off = 16; off > 0; off >>= 1) ss += __shfl_xor(ss, off, 32);
    float denom = fmaxf(sqrtf(ss), 1e-12f);
    float* o = out + (size_t)wave * 64 + lane * 2;
    o[0] = e0 / denom;
    o[1] = e1 / denom;
}

// ---------------------------------------------------------------------------
extern "C" void kernel_launch(void* const* d_in, const int* in_sizes, int n_in,
                              void* d_out, int out_size, void* d_ws, size_t ws_size,
                              hipStream_t stream) {
    const float* x  = (const float*)d_in[0];
    const int*   ei = (const int*)d_in[1];
    const float* W1 = (const float*)d_in[2];
    const float* b1 = (const float*)d_in[3];
    const float* W2 = (const float*)d_in[4];
    const float* b2 = (const float*)d_in[5];
    const float* W3 = (const float*)d_in[6];
    const float* b3 = (const float*)d_in[7];
    float* out = (float*)d_out;

    const int N = in_sizes[0] / 64;
    const int E = in_sizes[1] / 2;
    const int* row = ei;        // edge_index[0] = source
    const int* col = ei + E;    // edge_index[1] = target

    // Workspace: dinv[N] | buf0[N*64] | buf1[N*64]; d_out doubles as buf2.
    float* dinv = (float*)d_ws;
    float* buf0 = dinv + N;
    float* buf1 = buf0 + (size_t)N * 64;

    const int T = 256;
    // degrees -> dinv
    k_set1<<<(N + T - 1) / T, T, 0, stream>>>(dinv, N);
    k_deg <<<(E + T - 1) / T, T, 0, stream>>>(col, dinv, E);
    k_dinv<<<(N + T - 1) / T, T, 0, stream>>>(dinv, N);

    auto gemm = [&](const float* X, const float* W, float* H, int K, int M) {
        int tiles = (N >> 4) * (M >> 4);
        if (tiles > 0)
            k_gemm_f32_wmma<<<(tiles + 3) / 4, 128, 0, stream>>>(X, W, H, N, K, M);
        int rowStart = (N >> 4) << 4;
        if (rowStart < N) {
            int cnt = (N - rowStart) * M;
            k_gemm_tail<<<(cnt + T - 1) / T, T, 0, stream>>>(X, W, H, rowStart, N, K, M);
        }
    };
    auto aggregate = [&](const float* h, float* acc, int d) {
        size_t nd = (size_t)N * d;
        k_selfloop<<<(int)((nd + T - 1) / T), T, 0, stream>>>(h, dinv, acc, N, d);
        size_t tot = (size_t)E * (d >> 2);
        k_scatter<<<(int)((tot + T - 1) / T), T, 0, stream>>>(h, row, col, dinv, acc, E, d);
    };

    // Layer 1: 64 -> 32
    gemm(x, W1, buf0, 64, 32);
    aggregate(buf0, buf1, 32);
    k_bias_relu<<<(int)(((size_t)N * 32 + T - 1) / T), T, 0, stream>>>(buf1, b1, buf0, N, 32);

    // Layer 2: 32 -> 64
    gemm(buf0, W2, buf1, 32, 64);
    aggregate(buf1, out, 64);
    k_bias_relu<<<(int)(((size_t)N * 64 + T - 1) / T), T, 0, stream>>>(out, b2, buf0, N, 64);

    // Layer 3: 64 -> 64, then L2 row-normalize (in-place safe: per-row)
    gemm(buf0, W3, buf1, 64, 64);
    aggregate(buf1, out, 64);
    k_l2norm<<<(N + 3) / 4, 128, 0, stream>>>(out, b3, out, N);
}